// AdditiveAttention_18116172054933
// MI455X (gfx1250) — compile-verified
//
#include <hip/hip_runtime.h>
#include <math.h>

// ---------------------------------------------------------------------------
// Additive (Bahdanau) attention for MI455X / gfx1250, wave32.
//   B=4, S=512, H=128
// Stage 1: q_proj/k_proj GEMMs  -> V_WMMA_F32_16X16X4_F32
// Stage 2: scores = v . tanh(qp+kp), softmax -> v_tanh_f32 / v_exp_f32 (VALU)
// Stage 3: out = aw @ value GEMM -> V_WMMA_F32_16X16X4_F32
// ---------------------------------------------------------------------------

#define Bz 4
#define Sz 512
#define Hz 128

typedef __attribute__((ext_vector_type(2))) float v2f;
typedef __attribute__((ext_vector_type(8))) float v8f;

#if __has_builtin(__builtin_amdgcn_tanhf)
#define TANHF(x) __builtin_amdgcn_tanhf(x)
#else
#define TANHF(x) tanhf(x)
#endif

// ---------------------------------------------------------------------------
// Kernel 1: projections.
//   q_proj[m, o] = sum_h query[m,h] * attn_W[o, h]       + attn_b[o]
//   k_proj[m, o] = sum_h key[m,h]   * attn_W[o, H + h]
// M = B*S = 2048 rows, N = H = 128, K = H = 128.
// One wave (32 threads) computes one 16x16 tile; K-loop of 32 WMMAs.
// gridDim.x = 2 * (2048/16) * (128/16) = 2048 blocks of 32 threads.
//
// VGPR fragment layouts (ISA 7.12.2, wave32):
//   A 16x4 f32 : lanes 0-15 hold {K=0,K=1} for M=lane; lanes 16-31 hold {K=2,K=3}
//   B 4x16 f32 : vgpr0 = rows K=0 (lanes 0-15) / K=1 (lanes 16-31); vgpr1 = K=2/K=3
//   C 16x16 f32: vgpr r = row M=r (lanes 0-15) / M=r+8 (lanes 16-31), N=lane&15
// ---------------------------------------------------------------------------
__global__ void proj_wmma_kernel(const float* __restrict__ query,
                                 const float* __restrict__ key,
                                 const float* __restrict__ attn_W,
                                 const float* __restrict__ attn_b,
                                 float* __restrict__ qp,
                                 float* __restrict__ kp) {
  const int lane  = threadIdx.x;        // 0..31
  const int id    = blockIdx.x;         // 0..2047
  const int proj  = id >> 10;           // 0 = q, 1 = k
  const int tile  = id & 1023;
  const int m0    = (tile >> 3) * 16;   // 128 M-tiles
  const int n0    = (tile & 7) * 16;    // 8 N-tiles
  const int khalf = lane >> 4;          // 0 or 1
  const int ln    = lane & 15;

  const float* X = proj ? key : query;                    // [2048 x 128]
  float*       O = proj ? kp  : qp;                       // [2048 x 128]
  const int wofs = proj ? Hz : 0;                         // column offset in attn_W (rows of 2H)

  const float* arow = X + (m0 + ln) * Hz + 2 * khalf;     // A[M=ln(+16), K=k+2*khalf+{0,1}]
  const float* brow = attn_W + (n0 + ln) * (2 * Hz) + wofs + khalf; // Bmat[k+khalf, n=ln]

  v8f acc = {};
  #pragma unroll 8
  for (int k = 0; k < Hz; k += 4) {
    v2f a = *(const v2f*)(arow + k);    // 8B aligned: row base is 512B aligned, +0/+8B
    v2f bb;
    bb.x = brow[k];                     // Bmat[k + khalf,     n]
    bb.y = brow[k + 2];                 // Bmat[k + 2 + khalf, n]
    acc = __builtin_amdgcn_wmma_f32_16x16x4_f32(false, a, false, bb,
                                                (short)0, acc, false, false);
  }

  const float bias = (proj == 0) ? attn_b[n0 + ln] : 0.0f;
  #pragma unroll
  for (int r = 0; r < 8; ++r) {
    const int m = m0 + r + 8 * khalf;
    O[m * Hz + n0 + ln] = acc[r] + bias;
  }
}

// ---------------------------------------------------------------------------
// Kernel 2: scores + softmax.
// One 256-thread block per (b, s). Each thread owns t = tid and t = tid+256.
//   score[t] = sum_h v[h] * tanh(qp[b,s,h] + kp[b,t,h])
// qp row and v_w staged in LDS (broadcast reads). kp read as float4 from L2
// (1 MB total, fully L2-resident; the kernel is tanh-throughput bound anyway).
// ---------------------------------------------------------------------------
__global__ void scores_softmax_kernel(const float* __restrict__ qp,
                                      const float* __restrict__ kp,
                                      const float* __restrict__ vw,
                                      float* __restrict__ aw /* [B,S,S] */) {
  const int s   = blockIdx.x;   // 0..511
  const int b   = blockIdx.y;   // 0..3
  const int tid = threadIdx.x;  // 0..255

  __shared__ float sh_q[Hz];
  __shared__ float sh_v[Hz];
  __shared__ float red[256];

  if (tid < Hz) {
    sh_q[tid] = qp[(b * Sz + s) * Hz + tid];
    sh_v[tid] = vw[tid];
  }
  __syncthreads();

  const float* kpb = kp + b * Sz * Hz;
  float sc[2];
  #pragma unroll
  for (int j = 0; j < 2; ++j) {
    const int t = tid + j * 256;
    const float* krow = kpb + t * Hz;
    float acc = 0.0f;
    for (int h = 0; h < Hz; h += 4) {
      const float4 kv = *(const float4*)(krow + h);
      acc += sh_v[h + 0] * TANHF(sh_q[h + 0] + kv.x);
      acc += sh_v[h + 1] * TANHF(sh_q[h + 1] + kv.y);
      acc += sh_v[h + 2] * TANHF(sh_q[h + 2] + kv.z);
      acc += sh_v[h + 3] * TANHF(sh_q[h + 3] + kv.w);
    }
    sc[j] = acc;
  }

  // block max-reduce
  red[tid] = fmaxf(sc[0], sc[1]);
  __syncthreads();
  for (int off = 128; off > 0; off >>= 1) {
    if (tid < off) red[tid] = fmaxf(red[tid], red[tid + off]);
    __syncthreads();
  }
  const float m = red[0];
  __syncthreads();

  const float e0 = __expf(sc[0] - m);
  const float e1 = __expf(sc[1] - m);

  // block sum-reduce
  red[tid] = e0 + e1;
  __syncthreads();
  for (int off = 128; off > 0; off >>= 1) {
    if (tid < off) red[tid] += red[tid + off];
    __syncthreads();
  }
  const float inv = 1.0f / red[0];

  float* awrow = aw + (b * Sz + s) * Sz;
  awrow[tid]       = e0 * inv;
  awrow[tid + 256] = e1 * inv;
}

// ---------------------------------------------------------------------------
// Kernel 3: output[b] = aw[b] (512x512) @ value[b] (512x128), fp32 WMMA.
// One wave per 16x16 tile, 128 WMMAs along K=512.
// grid = (S/16, H/16, B) = (32, 8, 4), block = 32 (one wave, EXEC all-1s).
// ---------------------------------------------------------------------------
__global__ void av_wmma_kernel(const float* __restrict__ aw,
                               const float* __restrict__ value,
                               float* __restrict__ out) {
  const int lane  = threadIdx.x;
  const int m0    = blockIdx.x * 16;
  const int n0    = blockIdx.y * 16;
  const int b     = blockIdx.z;
  const int khalf = lane >> 4;
  const int ln    = lane & 15;

  const float* A = aw + b * Sz * Sz;      // [512 x 512]
  const float* V = value + b * Sz * Hz;   // [512 x 128]

  const float* arow = A + (m0 + ln) * Sz + 2 * khalf;
  v8f acc = {};
  #pragma unroll 8
  for (int k = 0; k < Sz; k += 4) {
    v2f a = *(const v2f*)(arow + k);            // aw[b, m, k+2*khalf+{0,1}]
    v2f bb;
    bb.x = V[(k + khalf) * Hz + n0 + ln];       // Bmat[k+khalf,     n]
    bb.y = V[(k + 2 + khalf) * Hz + n0 + ln];   // Bmat[k+2+khalf,   n]
    acc = __builtin_amdgcn_wmma_f32_16x16x4_f32(false, a, false, bb,
                                                (short)0, acc, false, false);
  }

  #pragma unroll
  for (int r = 0; r < 8; ++r) {
    const int m = m0 + r + 8 * khalf;
    out[(b * Sz + m) * Hz + n0 + ln] = acc[r];
  }
}

// ---------------------------------------------------------------------------
extern "C" void kernel_launch(void* const* d_in, const int* in_sizes, int n_in,
                              void* d_out, int out_size, void* d_ws, size_t ws_size,
                              hipStream_t stream) {
  const float* query  = (const float*)d_in[0];  // [B,S,H]
  const float* key    = (const float*)d_in[1];  // [B,S,H]
  const float* value  = (const float*)d_in[2];  // [B,S,H]
  const float* attn_W = (const float*)d_in[3];  // [H,2H]
  const float* attn_b = (const float*)d_in[4];  // [H]
  const float* v_w    = (const float*)d_in[5];  // [1,H]

  float* out = (float*)d_out;                   // output: B*S*H floats
  float* aw  = out + Bz * Sz * Hz;              // attention_weights: B*S*S floats

  float* qp = (float*)d_ws;                     // [B*S, H]
  float* kp = qp + Bz * Sz * Hz;                // [B*S, H]

  // Stage 1: projections (2 * 128 M-tiles * 8 N-tiles waves)
  proj_wmma_kernel<<<2048, 32, 0, stream>>>(query, key, attn_W, attn_b, qp, kp);

  // Stage 2: scores + softmax, one block per (b,s)
  dim3 g2(Sz, Bz);
  scores_softmax_kernel<<<g2, 256, 0, stream>>>(qp, kp, v_w, aw);

  // Stage 3: aw @ value
  dim3 g3(Sz / 16, Hz / 16, Bz);
  av_wmma_kernel<<<g3, 32, 0, stream>>>(aw, value, out);
}